// SMPLNN_12463995093356
// MI455X (gfx1250) — compile-verified
//
#include <hip/hip_runtime.h>
#include <hip/hip_bf16.h>

#define N_PTS      100000
#define N_VERTS    6890
#define N_JOINTS   24
#define N_TILES    ((N_VERTS + 15) / 16)   // 431 real vertex tiles of 16
#define N_TILES_RT 432                     // even compute trip count
#define PF_DIST    8                       // prefetch distance in tiles (2 KB)
#define N_TILES_PD (N_TILES_RT + PF_DIST)  // 440: buffer incl. prefetch pad

typedef __attribute__((ext_vector_type(2))) float v2f;
typedef __attribute__((ext_vector_type(8))) float v8f;

// ---------------------------------------------------------------------------
// Kernel 0: pre-swizzle smpl_verts into the WMMA B-operand (4x16 f32) lane
// layout for V_WMMA_F32_16X16X4_F32:
//   lanes  0-15 : VGPR0 = B[K=0][n]=v.x, VGPR1 = B[K=1][n]=v.y   (n = lane)
//   lanes 16-31 : VGPR0 = B[K=2][n]=v.z, VGPR1 = B[K=3][n]=-0.5*|v|^2
// With A rows [x,y,z,1], D[m][n] = x.v - 0.5|v|^2, so argmax_n D == argmin d2.
// Tail / pad columns get bias -1e30 so they can never win; dead tiles pad the
// buffer so the prefetched loop needs no edge branches.
// ---------------------------------------------------------------------------
__global__ void SMPLNN_pack_b(const float* __restrict__ sv,
                              float2* __restrict__ bpack) {
    int gid = blockIdx.x * blockDim.x + threadIdx.x;
    if (gid >= N_TILES_PD * 32) return;
    int t  = gid >> 5;
    int l  = gid & 31;
    int vi = t * 16 + (l & 15);
    float2 o;
    if (vi < N_VERTS) {
        float x = sv[vi * 3 + 0];
        float y = sv[vi * 3 + 1];
        float z = sv[vi * 3 + 2];
        if (l < 16) { o.x = x; o.y = y; }
        else        { o.x = z; o.y = -0.5f * (x * x + y * y + z * z); }
    } else {
        o.x = 0.0f;
        o.y = (l < 16) ? 0.0f : -1.0e30f;
    }
    bpack[gid] = o;
}

// ---------------------------------------------------------------------------
// Kernel 1: 8 waves / block, each wave owns 32 points (two 16-row A tiles).
// Inner loop: 2 B-tiles and 4 WMMAs per iteration; global_prefetch_b8 runs
// 8 tiles ahead (no LOADcnt, no register dep -> compiler cannot collapse it),
// so the blocking global_load_b64 hits a warm near cache and its
// s_wait_loadcnt is ~free. Phase 2: all 256 threads, one point each.
// ---------------------------------------------------------------------------
__global__ __launch_bounds__(256) void SMPLNN_main(
        const float* __restrict__ xyz,
        const float* __restrict__ rot,
        const float* __restrict__ sw,
        const float* __restrict__ tm,
        const float2* __restrict__ bpack,
        float* __restrict__ out) {
    __shared__ float ldsT[N_JOINTS * 16];   // 24 x 4x4 transforms
    __shared__ int   ldsIdx[256];           // winning vertex per point

    const int tid       = threadIdx.x;
    const int lane      = tid & 31;
    const int wave      = tid >> 5;
    const int blockBase = blockIdx.x * 256;

    // cache transforms_mat in LDS for phase 2
    for (int i = tid; i < N_JOINTS * 16; i += 256) ldsT[i] = tm[i];

    const int waveBase = blockBase + wave * 32;   // wave-uniform
    if (waveBase < N_PTS) {                       // uniform guard: EXEC all-1s
        const int col  = lane & 15;
        const int row0 = waveBase + col;          // N_PTS % 32 == 0: all valid
        const int row1 = waveBase + 16 + col;

        // A operands: lanes 0-15 hold {x,y}; lanes 16-31 hold {z,1}
        v2f a0, a1;
        if (lane < 16) {
            a0.x = xyz[row0 * 3 + 0]; a0.y = xyz[row0 * 3 + 1];
            a1.x = xyz[row1 * 3 + 0]; a1.y = xyz[row1 * 3 + 1];
        } else {
            a0.x = xyz[row0 * 3 + 2]; a0.y = 1.0f;
            a1.x = xyz[row1 * 3 + 2]; a1.y = 1.0f;
        }

        float best0[8], best1[8];
        int   bidx0[8], bidx1[8];
#pragma unroll
        for (int r = 0; r < 8; ++r) {
            best0[r] = -3.0e38f; bidx0[r] = 0;
            best1[r] = -3.0e38f; bidx1[r] = 0;
        }

        const float2* bp = bpack + lane;
        for (int t = 0; t < N_TILES_RT; t += 2) {
            // run ahead of the blocking loads; no LOADcnt, no reg dependence
            __builtin_prefetch(bp + (t + PF_DIST) * 32, 0, 3);
            __builtin_prefetch(bp + (t + PF_DIST + 1) * 32, 0, 3);

            float2 bAr = bp[t * 32];
            float2 bBr = bp[(t + 1) * 32];

            // ---- tile t ----
            {
                v2f b; b.x = bAr.x; b.y = bAr.y;
                v8f c0 = {}, c1 = {};
                c0 = __builtin_amdgcn_wmma_f32_16x16x4_f32(
                        false, a0, false, b, (short)0, c0, false, false);
                c1 = __builtin_amdgcn_wmma_f32_16x16x4_f32(
                        false, a1, false, b, (short)0, c1, false, false);
                const int idx = t * 16 + col;
#pragma unroll
                for (int r = 0; r < 8; ++r) {
                    // strictly-greater keeps the earliest index on ties
                    if (c0[r] > best0[r]) { best0[r] = c0[r]; bidx0[r] = idx; }
                    if (c1[r] > best1[r]) { best1[r] = c1[r]; bidx1[r] = idx; }
                }
            }
            // ---- tile t+1 ----
            {
                v2f b; b.x = bBr.x; b.y = bBr.y;
                v8f c0 = {}, c1 = {};
                c0 = __builtin_amdgcn_wmma_f32_16x16x4_f32(
                        false, a0, false, b, (short)0, c0, false, false);
                c1 = __builtin_amdgcn_wmma_f32_16x16x4_f32(
                        false, a1, false, b, (short)0, c1, false, false);
                const int idx = (t + 1) * 16 + col;
#pragma unroll
                for (int r = 0; r < 8; ++r) {
                    if (c0[r] > best0[r]) { best0[r] = c0[r]; bidx0[r] = idx; }
                    if (c1[r] > best1[r]) { best1[r] = c1[r]; bidx1[r] = idx; }
                }
            }
        }

        // Tournament over the 16-lane half (masks 1,2,4,8 never cross halves).
        // Tie-break toward the smaller index to match jnp.argmin semantics.
#pragma unroll
        for (int m = 8; m >= 1; m >>= 1) {
#pragma unroll
            for (int r = 0; r < 8; ++r) {
                float os0 = __shfl_xor(best0[r], m, 32);
                int   oi0 = __shfl_xor(bidx0[r], m, 32);
                if (os0 > best0[r] || (os0 == best0[r] && oi0 < bidx0[r])) {
                    best0[r] = os0; bidx0[r] = oi0;
                }
                float os1 = __shfl_xor(best1[r], m, 32);
                int   oi1 = __shfl_xor(bidx1[r], m, 32);
                if (os1 > best1[r] || (os1 == best1[r] && oi1 < bidx1[r])) {
                    best1[r] = os1; bidx1[r] = oi1;
                }
            }
        }
        // D layout: VGPR r -> M=r (lanes 0-15) / M=r+8 (lanes 16-31)
        const int wloc = wave * 32;
        if (lane == 0) {
#pragma unroll
            for (int r = 0; r < 8; ++r) {
                ldsIdx[wloc + r]      = bidx0[r];
                ldsIdx[wloc + 16 + r] = bidx1[r];
            }
        } else if (lane == 16) {
#pragma unroll
            for (int r = 0; r < 8; ++r) {
                ldsIdx[wloc + 8 + r]  = bidx0[r];
                ldsIdx[wloc + 24 + r] = bidx1[r];
            }
        }
    }
    __syncthreads();

    // ---------------- Phase 2: per-point epilogue -----------------
    {
        const int p = blockBase + tid;
        if (p < N_PTS) {
            const int j = ldsIdx[tid];

            // T_fwd = skinning_weights[j] @ transforms (24 x 16 blend)
            float T[16];
#pragma unroll
            for (int i = 0; i < 16; ++i) T[i] = 0.0f;
            const float* w = sw + j * N_JOINTS;
            for (int k = 0; k < N_JOINTS; ++k) {
                const float wk = w[k];
#pragma unroll
                for (int i = 0; i < 16; ++i) T[i] = fmaf(wk, ldsT[k * 16 + i], T[i]);
            }

            const float x = xyz[p * 3 + 0];
            const float y = xyz[p * 3 + 1];
            const float z = xyz[p * 3 + 2];
            const float xb0 = T[0] * x + T[1] * y + T[2]  * z + T[3];
            const float xb1 = T[4] * x + T[5] * y + T[6]  * z + T[7];
            const float xb2 = T[8] * x + T[9] * y + T[10] * z + T[11];

            // quaternion -> rotation matrix
            float qr = rot[p * 4 + 0], qx = rot[p * 4 + 1];
            float qy = rot[p * 4 + 2], qz = rot[p * 4 + 3];
            const float inv = 1.0f / sqrtf(qr * qr + qx * qx + qy * qy + qz * qz);
            qr *= inv; qx *= inv; qy *= inv; qz *= inv;
            float R[9];
            R[0] = 1.0f - 2.0f * (qy * qy + qz * qz);
            R[1] = 2.0f * (qx * qy - qr * qz);
            R[2] = 2.0f * (qx * qz + qr * qy);
            R[3] = 2.0f * (qx * qy + qr * qz);
            R[4] = 1.0f - 2.0f * (qx * qx + qz * qz);
            R[5] = 2.0f * (qy * qz - qr * qx);
            R[6] = 2.0f * (qx * qz - qr * qy);
            R[7] = 2.0f * (qy * qz + qr * qx);
            R[8] = 1.0f - 2.0f * (qx * qx + qy * qy);

            float* xbar = out;                          // N_PTS*3
            float* rbar = out + N_PTS * 3;              // N_PTS*9
            float* tfwd = out + N_PTS * 3 + N_PTS * 9;  // N_PTS*16

            xbar[p * 3 + 0] = xb0;
            xbar[p * 3 + 1] = xb1;
            xbar[p * 3 + 2] = xb2;
#pragma unroll
            for (int i = 0; i < 3; ++i)
#pragma unroll
                for (int kk = 0; kk < 3; ++kk)
                    rbar[p * 9 + i * 3 + kk] =
                        T[i * 4 + 0] * R[0 * 3 + kk] +
                        T[i * 4 + 1] * R[1 * 3 + kk] +
                        T[i * 4 + 2] * R[2 * 3 + kk];
#pragma unroll
            for (int i = 0; i < 16; ++i) tfwd[p * 16 + i] = T[i];
        }
    }
}

extern "C" void kernel_launch(void* const* d_in, const int* in_sizes, int n_in,
                              void* d_out, int out_size, void* d_ws, size_t ws_size,
                              hipStream_t stream) {
    const float* xyz = (const float*)d_in[0];   // (100000,3)
    const float* rot = (const float*)d_in[1];   // (100000,4)
    const float* sv  = (const float*)d_in[2];   // (6890,3)
    const float* sw  = (const float*)d_in[3];   // (6890,24)
    const float* tm  = (const float*)d_in[4];   // (24,4,4)
    float*       out = (float*)d_out;           // x_bar | rotation_bar | T_fwd

    float2* bpack = (float2*)d_ws;              // N_TILES_PD*32 float2 ~ 113 KB

    SMPLNN_pack_b<<<(N_TILES_PD * 32 + 255) / 256, 256, 0, stream>>>(sv, bpack);
    SMPLNN_main<<<(N_PTS + 255) / 256, 256, 0, stream>>>(xyz, rot, sw, tm, bpack, out);
}